// CrossAttention_71176198029334
// MI455X (gfx1250) — compile-verified
//
#include <hip/hip_runtime.h>

typedef __attribute__((ext_vector_type(2))) float v2f;
typedef __attribute__((ext_vector_type(4))) float v4f;
typedef __attribute__((ext_vector_type(8))) float v8f;

constexpr int B  = 4;
constexpr int C  = 256;
constexpr int T  = 4096;
constexpr int U  = 64;
constexpr int AW = 256;            // attention width
constexpr int MT = T / 16;         // 256 row tiles per batch
constexpr int NTU = U / 16;        // 4 u-tiles
constexpr int JR = 8;              // +/- 8 j-tiles around the diagonal tile
constexpr int JS = 2 * JR + 1;     // 17 j-tiles cover the whole band window

static __device__ __forceinline__ v8f wmma_f32(v2f a, v2f b, v8f c) {
  return __builtin_amdgcn_wmma_f32_16x16x4_f32(false, a, false, b, (short)0, c,
                                               false, false);
}

// ---------------------------------------------------------------- zero fill a
// 256 MB streaming store > 192 MB L2: use non-temporal stores so the fill
// doesn't evict the workspace / band tiles that later kernels re-read.
__global__ void zero_a(v4f* __restrict__ a4, long n4) {
  long idx = (long)blockIdx.x * blockDim.x + threadIdx.x;
  long stride = (long)gridDim.x * blockDim.x;
  v4f z = {0.f, 0.f, 0.f, 0.f};
  for (long i = idx; i < n4; i += stride)
    __builtin_nontemporal_store(z, &a4[i]);
}

// ------------------------------------------------- projections q1,k2,v2 (ws)
// q1[b,t,u] = sum_c y1[b,c,t] * Wq[c,u]   (A is y^T: A(t,c)=y[b,c*T+t])
__global__ void proj(const float* __restrict__ y1, const float* __restrict__ y2,
                     const float* __restrict__ Wq, const float* __restrict__ Wk,
                     const float* __restrict__ Wv, float* __restrict__ ws) {
  int lane = threadIdx.x;
  int half = lane >> 4, l16 = lane & 15;
  int wid = blockIdx.x;                 // [3][B][MT][NTU]
  int nt = wid % NTU; wid /= NTU;
  int mt = wid % MT;  wid /= MT;
  int b  = wid % B;   wid /= B;
  int mat = wid;                        // 0=q, 1=k, 2=v
  const float* Y = (mat == 0) ? y1 : y2;
  const float* W = (mat == 0) ? Wq : (mat == 1) ? Wk : Wv;
  const float* Yb = Y + (size_t)b * C * T;
  float* Db = ws + ((size_t)mat * B + b) * T * U;

  int i0 = mt * 16, u0 = nt * 16;
  v8f acc = {};
  for (int k0 = 0; k0 < C; k0 += 4) {
    int ka = k0 + half * 2;             // A: VGPR0=K+0/K+2, VGPR1=K+1/K+3
    v2f af, bf;
    af.x = Yb[(size_t)(ka + 0) * T + i0 + l16];
    af.y = Yb[(size_t)(ka + 1) * T + i0 + l16];
    bf.x = W[(size_t)(ka + 0) * U + u0 + l16];
    bf.y = W[(size_t)(ka + 1) * U + u0 + l16];
    acc = wmma_f32(af, bf, acc);
  }
#pragma unroll
  for (int r = 0; r < 8; ++r) {         // C/D: lanes0-15 row r, lanes16-31 row r+8
    int m = r + half * 8;
    Db[(size_t)(i0 + m) * U + u0 + l16] = acc[r];
  }
}

// ------------------------------------ banded scores + softmax + store into a
__global__ void attn(const float* __restrict__ ws, float* __restrict__ a_out) {
  const float* q1 = ws;
  const float* k2 = ws + (size_t)B * T * U;
  int lane = threadIdx.x;
  int half = lane >> 4, l16 = lane & 15;
  int wid = blockIdx.x;                 // [B][MT]
  int mt = wid % MT, b = wid / MT;
  int i0 = mt * 16;
  const float* qb = q1 + (size_t)b * T * U;
  const float* kb = k2 + (size_t)b * T * U;

  // Q fragments are invariant across j-tiles: load once (16 k-steps x v2f).
  v2f qf[U / 4];
#pragma unroll
  for (int k0 = 0; k0 < U; k0 += 4) {
    int ka = k0 + half * 2;
    qf[k0 / 4].x = qb[(size_t)(i0 + l16) * U + ka + 0];
    qf[k0 / 4].y = qb[(size_t)(i0 + l16) * U + ka + 1];
  }

  v8f s[JS];
  int jt0 = mt - JR;
#pragma unroll
  for (int t = 0; t < JS; ++t) {
    v8f acc = {};
    int jt = jt0 + t;
    if (jt >= 0 && jt < MT) {           // uniform per wave: EXEC stays full
      int j0 = jt * 16;
      for (int k0 = 0; k0 < U; k0 += 4) {
        int ka = k0 + half * 2;
        v2f bf;                         // S = Q*K^T: B(u,n)=k[j0+n,u]
        bf.x = kb[(size_t)(j0 + l16) * U + ka + 0];
        bf.y = kb[(size_t)(j0 + l16) * U + ka + 1];
        acc = wmma_f32(qf[k0 / 4], bf, acc);
      }
    }
    s[t] = acc;
  }

  const float scale = 0.125f;           // 1/sqrt(64)
  float rmax[8];
#pragma unroll
  for (int r = 0; r < 8; ++r) rmax[r] = -3.0e38f;
#pragma unroll
  for (int t = 0; t < JS; ++t) {
    int jt = jt0 + t;
    int j0 = jt * 16;
#pragma unroll
    for (int r = 0; r < 8; ++r) {
      int i = i0 + r + half * 8;
      int j = j0 + l16;
      bool valid = (jt >= 0) && (jt < MT) && (j - AW / 2 <= i) &&
                   (i < j - AW / 2 + AW);
      float e = valid ? s[t][r] * scale : -3.0e38f;
      s[t][r] = e;
      rmax[r] = fmaxf(rmax[r], e);
    }
  }
#pragma unroll
  for (int r = 0; r < 8; ++r) {         // reduce within each 16-lane half (rows)
    float v = rmax[r];
    for (int m = 1; m < 16; m <<= 1) v = fmaxf(v, __shfl_xor(v, m, 32));
    rmax[r] = v;
  }
  float rsum[8];
#pragma unroll
  for (int r = 0; r < 8; ++r) rsum[r] = 0.f;
#pragma unroll
  for (int t = 0; t < JS; ++t) {
#pragma unroll
    for (int r = 0; r < 8; ++r) {
      float p = __expf(s[t][r] - rmax[r]);   // masked -> exp(-huge) = 0
      s[t][r] = p;
      rsum[r] += p;
    }
  }
#pragma unroll
  for (int r = 0; r < 8; ++r) {
    float v = rsum[r];
    for (int m = 1; m < 16; m <<= 1) v += __shfl_xor(v, m, 32);
    rsum[r] = 1.0f / v;
  }
  float* ab = a_out + (size_t)b * T * T;
#pragma unroll
  for (int t = 0; t < JS; ++t) {
    int jt = jt0 + t;
    if (jt < 0 || jt >= MT) continue;
    int j0 = jt * 16;
#pragma unroll
    for (int r = 0; r < 8; ++r) {       // regular-temporal: outk re-reads these
      int m = r + half * 8;
      ab[(size_t)(i0 + m) * T + j0 + l16] = s[t][r] * rsum[r];
    }
  }
}

// -------------------------------------- out[b,u,i] = sum_j a[b,i,j]*v2[b,j,u]
__global__ void outk(const float* __restrict__ ws, const float* __restrict__ a_in,
                     float* __restrict__ out) {
  const float* v2 = ws + (size_t)2 * B * T * U;
  int lane = threadIdx.x;
  int half = lane >> 4, l16 = lane & 15;
  int wid = blockIdx.x;                 // [B][MT][NTU]
  int nt = wid % NTU; wid /= NTU;
  int mt = wid % MT;  int b = wid / MT;
  int i0 = mt * 16, u0 = nt * 16;
  const float* ab = a_in + (size_t)b * T * T;
  const float* vb = v2 + (size_t)b * T * U;

  v8f acc = {};
  int jt0 = mt - JR;
  for (int t = 0; t < JS; ++t) {
    int jt = jt0 + t;
    if (jt < 0 || jt >= MT) continue;   // uniform per wave
    int j0 = jt * 16;
    for (int k0 = 0; k0 < 16; k0 += 4) {
      int ka = k0 + half * 2;
      v2f af, bf;                       // A(m,k)=a[i0+m, j0+k], B(k,n)=v[j0+k, u0+n]
      af.x = ab[(size_t)(i0 + l16) * T + j0 + ka + 0];
      af.y = ab[(size_t)(i0 + l16) * T + j0 + ka + 1];
      bf.x = vb[(size_t)(j0 + ka + 0) * U + u0 + l16];
      bf.y = vb[(size_t)(j0 + ka + 1) * U + u0 + l16];
      acc = wmma_f32(af, bf, acc);
    }
  }
  float* ob = out + (size_t)b * U * T;
#pragma unroll
  for (int r = 0; r < 8; ++r) {         // transposed store: out[b, u0+n, i0+m]
    int m = r + half * 8;
    ob[(size_t)(u0 + l16) * T + i0 + m] = acc[r];
  }
}

extern "C" void kernel_launch(void* const* d_in, const int* in_sizes, int n_in,
                              void* d_out, int out_size, void* d_ws, size_t ws_size,
                              hipStream_t stream) {
  const float* y1 = (const float*)d_in[0];
  const float* y2 = (const float*)d_in[1];
  const float* Wq = (const float*)d_in[2];
  const float* Wk = (const float*)d_in[3];
  const float* Wv = (const float*)d_in[4];
  float* out = (float*)d_out;                    // B*U*T floats, first output
  float* a   = out + (size_t)B * U * T;          // B*T*T floats, second output
  float* ws  = (float*)d_ws;                     // q1|k2|v2, 3*B*T*U floats (12 MB)

  long n4 = (long)B * T * T / 4;
  zero_a<<<2048, 256, 0, stream>>>((v4f*)a, n4);
  proj<<<3 * B * MT * NTU, 32, 0, stream>>>(y1, y2, Wq, Wk, Wv, ws);
  attn<<<B * MT, 32, 0, stream>>>(ws, a);
  outk<<<B * MT * NTU, 32, 0, stream>>>(ws, a, out);
}